// RPNPostProcess_29815662969435
// MI455X (gfx1250) — compile-verified
//
#include <hip/hip_runtime.h>
#include <hip/hip_bf16.h>
#include <math.h>

// ---------------------------------------------------------------------------
// RPN post-process for MI455X (gfx1250, wave32).
// Pipeline: decode -> radix-histogram top-k select -> LDS bitonic sort ->
//           WMMA (f32 16x16x4) corner projection (+fused standup reduction) ->
//           IoU mask -> serial NMS with row prefetch.
// Memory-bound workload (~50MB traffic ~ 2us HBM floor @ 23.3 TB/s); the only
// dense linear algebra (4x4 projective transform of 32768 corner points) is
// mapped onto V_WMMA_F32_16X16X4_F32, 4 unrolled WMMAs per wave (8 boxes).
// ---------------------------------------------------------------------------

#define Hh 256
#define Ww 512
#define HW (Hh*Ww)          // 131072
#define NHALF (HW*2)        // 262144 boxes per branch
#define NTOT (NHALF*2)      // 524288 boxes total
#define TOPK 4096
#define SCORE_THR 0.2f
#define NMS_THR 0.15f
#define NBINS 65536
#define CAND_CAP 8192

// ---- workspace layout (32-bit word offsets into d_ws) ----
#define HIST_OFF   0u                         // 65536 u32
#define CNT_OFF    65536u                     // 1 u32
#define BIN_OFF    65537u                     // 1 u32
#define CAND_OFF   65540u                     // 8192 u64 (byte off 262160, 8-aligned)
#define UKEY_OFF   81924u                     // NTOT u32
#define BOXES_OFF  (UKEY_OFF + NTOT)          // NTOT*7 f32
#define TOPI_OFF   (BOXES_OFF + NTOT*7u)      // 4096 u32
#define TOPS_OFF   (TOPI_OFF + TOPK)          // 4096 f32
#define STAND_OFF  (TOPS_OFF + TOPK)          // 4096*4 f32
#define MASK_OFF   (STAND_OFF + TOPK*4u)      // 4096*128 u32
#define ZERO_WORDS 81924u                     // hist + cnt + bin + cand region

typedef float v2f __attribute__((ext_vector_type(2)));
typedef float v8f __attribute__((ext_vector_type(8)));

// ---------------------------------------------------------------------------
__global__ void k_zero(unsigned* ws) {
    unsigned n = blockIdx.x * blockDim.x + threadIdx.x;
    if (n < ZERO_WORDS) ws[n] = 0u;
}

// ---------------------------------------------------------------------------
// Fused decode: sigmoid score, delta->box, branch-i reprojection via t_proj
// + corner_to_center, sortable key, 16-bit-radix histogram (one global pass).
__global__ void k_decode(const float* __restrict__ anchors,
                         const float* __restrict__ psm_v, const float* __restrict__ rm_v,
                         const float* __restrict__ psm_i, const float* __restrict__ rm_i,
                         const float* __restrict__ t_proj,
                         float* __restrict__ boxes, unsigned* __restrict__ ukeys,
                         unsigned* __restrict__ hist) {
    int n = blockIdx.x * blockDim.x + threadIdx.x;
    if (n >= NTOT) return;
    int br = (n >= NHALF) ? 1 : 0;
    int m  = n - br * NHALF;          // row of (H*W*2, 7)
    int a  = m & 1;                   // anchor id
    int hw = m >> 1;                  // h*W + w
    const float* psm = br ? psm_i : psm_v;
    const float* rm  = br ? rm_i  : rm_v;

    float s = 1.0f / (1.0f + expf(-psm[a * HW + hw]));

    float d[7], an[7];
    #pragma unroll
    for (int j = 0; j < 7; ++j) {
        d[j]  = rm[(a * 7 + j) * HW + hw];
        an[j] = anchors[(size_t)m * 7 + j];
    }
    float ad = sqrtf(an[4] * an[4] + an[5] * an[5]);
    float bx = d[0] * ad + an[0];
    float by = d[1] * ad + an[1];
    float bz = d[2] * an[3] + an[2];
    float bh = expf(d[3]) * an[3];
    float bw = expf(d[4]) * an[4];
    float bl = expf(d[5]) * an[5];
    float byaw = d[6] + an[6];

    if (br == 1) {
        // boxes_to_corners_3d -> project(t_proj) -> corner_to_center
        float cx[8], cy[8], cz[8];
        float cs = cosf(byaw), sn = sinf(byaw);
        #pragma unroll
        for (int c = 0; c < 8; ++c) {
            float sx = (c & 2) ? -0.5f : 0.5f;
            float sy = (((c ^ (c >> 1)) & 1) ? -0.5f : 0.5f);
            float sz = (c & 4) ? 0.5f : -0.5f;
            float lx = bl * sx, ly = bw * sy, lz = bh * sz;
            float px = lx * cs - ly * sn + bx;
            float py = lx * sn + ly * cs + by;
            float pz = lz + bz;
            cx[c] = t_proj[0] * px + t_proj[1] * py + t_proj[2]  * pz + t_proj[3];
            cy[c] = t_proj[4] * px + t_proj[5] * py + t_proj[6]  * pz + t_proj[7];
            cz[c] = t_proj[8] * px + t_proj[9] * py + t_proj[10] * pz + t_proj[11];
        }
        float mx = 0.f, my = 0.f, mz = 0.f;
        #pragma unroll
        for (int c = 0; c < 8; ++c) { mx += cx[c]; my += cy[c]; mz += cz[c]; }
        mx *= 0.125f; my *= 0.125f; mz *= 0.125f;
        float nh = 0.25f * (cz[4] + cz[5] + cz[6] + cz[7]) -
                   0.25f * (cz[0] + cz[1] + cz[2] + cz[3]);
        const int la[4] = {0, 1, 4, 5}, lb[4] = {3, 2, 7, 6};
        const int wa[4] = {0, 3, 4, 7}, wb[4] = {1, 2, 5, 6};
        float nl = 0.f, nw = 0.f, dsx = 0.f, dsy = 0.f;
        #pragma unroll
        for (int k = 0; k < 4; ++k) {
            float dx = cx[la[k]] - cx[lb[k]], dy = cy[la[k]] - cy[lb[k]];
            nl += sqrtf(dx * dx + dy * dy);
            dsx += dx; dsy += dy;
            float ex = cx[wa[k]] - cx[wb[k]], ey = cy[wa[k]] - cy[wb[k]];
            nw += sqrtf(ex * ex + ey * ey);
        }
        nl *= 0.25f; nw *= 0.25f;
        bx = mx; by = my; bz = mz; bh = nh; bw = nw; bl = nl;
        byaw = atan2f(dsy, dsx);
    }
    float* bp = boxes + (size_t)n * 7;
    bp[0] = bx; bp[1] = by; bp[2] = bz; bp[3] = bh; bp[4] = bw; bp[5] = bl; bp[6] = byaw;

    float masked = (s > SCORE_THR) ? s : -1.0f;
    unsigned kb = __float_as_uint(masked);
    unsigned uk = (kb & 0x80000000u) ? ~kb : (kb ^ 0x80000000u); // monotone
    ukeys[n] = uk;
    atomicAdd(&hist[uk >> 16], 1u);
}

// ---------------------------------------------------------------------------
// Find boundary bin B (descending) such that count(bin >= B) first reaches TOPK.
__global__ void k_selectbin(const unsigned* __restrict__ hist, unsigned* __restrict__ selBin) {
    __shared__ unsigned part[256];
    int t = threadIdx.x;
    unsigned s = 0;
    int hiBin = NBINS - 1 - t * 256;
    for (int r = 0; r < 256; ++r) s += hist[hiBin - r];
    part[t] = s;
    __syncthreads();
    if (t == 0) {
        unsigned cum = 0, res = 0;
        for (int c = 0; c < 256; ++c) {
            if (cum + part[c] >= TOPK) {
                int hb = NBINS - 1 - c * 256;
                for (int r = 0; r < 256; ++r) {
                    cum += hist[hb - r];
                    if (cum >= TOPK) { res = (unsigned)(hb - r); break; }
                }
                *selBin = res;
                return;
            }
            cum += part[c];
        }
        *selBin = 0u;
    }
}

// ---------------------------------------------------------------------------
// Compact candidates (bin >= B) as 64-bit keys (score_key << 32) | ~index.
__global__ void k_compact(const unsigned* __restrict__ ukeys,
                          const unsigned* __restrict__ selBin,
                          unsigned* __restrict__ cnt,
                          unsigned long long* __restrict__ cand) {
    int n = blockIdx.x * blockDim.x + threadIdx.x;
    if (n >= NTOT) return;
    unsigned u = ukeys[n];
    if ((u >> 16) >= *selBin) {
        unsigned pos = atomicAdd(cnt, 1u);
        if (pos < CAND_CAP)
            cand[pos] = ((unsigned long long)u << 32) |
                        (unsigned long long)(0xFFFFFFFFu - (unsigned)n);
    }
}

// ---------------------------------------------------------------------------
// One-workgroup bitonic sort of 8192 candidate keys in LDS (64KB of the 320KB
// WGP pool), descending; emits exact jax.lax.top_k order (ties -> low index).
__global__ void __launch_bounds__(1024) k_sortcand(const unsigned long long* __restrict__ cand,
                                                   unsigned* __restrict__ topi,
                                                   float* __restrict__ tops) {
    __shared__ unsigned long long sk[CAND_CAP];
    int tid = threadIdx.x;
    for (int i = tid; i < CAND_CAP; i += 1024) sk[i] = cand[i];
    __syncthreads();
    for (unsigned k = 2; k <= CAND_CAP; k <<= 1) {
        for (unsigned j = k >> 1; j > 0; j >>= 1) {
            for (unsigned i = tid; i < CAND_CAP; i += 1024) {
                unsigned ixj = i ^ j;
                if (ixj > i) {
                    unsigned long long a = sk[i], b = sk[ixj];
                    bool desc = ((i & k) == 0);
                    if ((a < b) == desc) { sk[i] = b; sk[ixj] = a; }
                }
            }
            __syncthreads();
        }
    }
    for (int i = tid; i < TOPK; i += 1024) {
        unsigned long long key = sk[i];
        if (key == 0ULL) { topi[i] = 0u; tops[i] = -1.0f; continue; }
        unsigned hi = (unsigned)(key >> 32);
        unsigned lo = (unsigned)key;
        unsigned sb = (hi & 0x80000000u) ? (hi ^ 0x80000000u) : ~hi;
        topi[i] = 0xFFFFFFFFu - lo;
        tops[i] = __uint_as_float(sb);
    }
}

// ---------------------------------------------------------------------------
// Corner generation + t_ego projection via V_WMMA_F32_16X16X4_F32.
//   A (16x4, K=4): 16 corner points as homogeneous (x,y,z,1) rows.
//     ISA layout: VGPR0 = K0 (lanes 0-15) / K2 (lanes 16-31); VGPR1 = K1 / K3.
//   B (4x16): B[k][n] = T_ego[n][k] for n<3 (VGPR0 = rows K0/K2 halves,
//     VGPR1 = rows K1/K3, mirroring the A layout); built once per wave.
//   D (16x16 f32): VGPR r holds M=r (lanes 0-15) / M=r+8 (lanes 16-31), N=lane%16.
// Each wave projects 8 boxes via 4 unrolled WMMAs; corners written straight
// into d_out rows of stride 25, and the standup (min/max x,y) reduction is
// fused here: lanes 0/1 (and 16/17) hold the full x/y columns of one box's
// 8 corners inside the D accumulator VGPRs.
__global__ void __launch_bounds__(256) k_project(const float* __restrict__ boxes,
                                                 const unsigned* __restrict__ topi,
                                                 const float* __restrict__ t_ego,
                                                 float* __restrict__ out,
                                                 float* __restrict__ standup) {
    int tid  = blockIdx.x * blockDim.x + threadIdx.x;
    int lane = tid & 31;
    int box0 = (tid >> 5) * 8;     // 512 waves x 8 boxes
    int p    = lane & 15;          // point id 0..15 within one WMMA
    int ncol = lane & 15;

    // B fragment: constant across the 4 WMMAs of this wave.
    v2f B;
    if (lane < 16) {
        B.x = (ncol < 3) ? t_ego[ncol * 4 + 0] : 0.0f;          // row K=0
        B.y = (ncol < 3) ? t_ego[ncol * 4 + 1] : 0.0f;          // row K=1
    } else {
        B.x = (ncol < 3) ? t_ego[ncol * 4 + 2] : 0.0f;          // row K=2
        B.y = (ncol < 3) ? t_ego[ncol * 4 + 3] : 0.0f;          // row K=3
    }

    #pragma unroll
    for (int it = 0; it < 4; ++it) {
        int boxBase = box0 + it * 2;
        int b = boxBase + (p >> 3);
        int c = p & 7;

        const float* bp = boxes + (size_t)topi[b] * 7;
        float bx = bp[0], by = bp[1], bz = bp[2];
        float bh = bp[3], bw = bp[4], bl = bp[5], byaw = bp[6];
        float sx = (c & 2) ? -0.5f : 0.5f;
        float sy = (((c ^ (c >> 1)) & 1) ? -0.5f : 0.5f);
        float sz = (c & 4) ? 0.5f : -0.5f;
        float lx = bl * sx, ly = bw * sy, lz = bh * sz;
        float cs = cosf(byaw), sn = sinf(byaw);
        float px = lx * cs - ly * sn + bx;
        float py = lx * sn + ly * cs + by;
        float pz = lz + bz;

        v2f A;
        if (lane < 16) { A.x = px; A.y = py; }   // K=0, K=1
        else           { A.x = pz; A.y = 1.0f; } // K=2, K=3

        v8f C = {0.f, 0.f, 0.f, 0.f, 0.f, 0.f, 0.f, 0.f};
        v8f D = __builtin_amdgcn_wmma_f32_16x16x4_f32(
            /*neg_a=*/false, A, /*neg_b=*/false, B,
            /*c_mod=*/(short)0, C, /*reuse_a=*/false, /*reuse_b=*/false);

        // corners -> d_out (stride-25 rows)
        if (lane < 3) {                 // coords x/y/z of box boxBase, corners 0..7
            #pragma unroll
            for (int r = 0; r < 8; ++r)
                out[(size_t)boxBase * 25 + r * 3 + lane] = D[r];
        } else if (lane >= 16 && lane < 19) {  // box boxBase+1
            int nc = lane - 16;
            #pragma unroll
            for (int r = 0; r < 8; ++r)
                out[(size_t)(boxBase + 1) * 25 + r * 3 + nc] = D[r];
        }

        // fused standup reduction: lanes 0/1 -> box boxBase, 16/17 -> boxBase+1
        int cl = lane & 15;
        if (cl < 2) {
            float mn = D[0], mx = D[0];
            #pragma unroll
            for (int r = 1; r < 8; ++r) { mn = fminf(mn, D[r]); mx = fmaxf(mx, D[r]); }
            int sbx = boxBase + (lane >> 4);   // 0/1 -> boxBase ; 16/17 -> boxBase+1
            standup[(size_t)sbx * 4 + cl]     = mn;   // min x (cl=0) / min y (cl=1)
            standup[(size_t)sbx * 4 + 2 + cl] = mx;   // max x / max y
        }
    }
}

// ---------------------------------------------------------------------------
// Suppression bitmask: mask[i][word] bit t set iff j=word*32+t > i and IoU>thr.
__global__ void __launch_bounds__(128) k_mask(const float* __restrict__ standup,
                                              unsigned* __restrict__ mask) {
    int i  = blockIdx.x;
    int wj = threadIdx.x;           // 128 words of 32 j's
    float x0 = standup[i * 4 + 0], y0 = standup[i * 4 + 1];
    float x1 = standup[i * 4 + 2], y1 = standup[i * 4 + 3];
    float ai = (x1 - x0) * (y1 - y0);
    unsigned bits = 0u;
    #pragma unroll 4
    for (int t = 0; t < 32; ++t) {
        int j = wj * 32 + t;
        if (j > i) {
            const float* sb = standup + (size_t)j * 4;
            float jx0 = sb[0], jy0 = sb[1], jx1 = sb[2], jy1 = sb[3];
            float aj = (jx1 - jx0) * (jy1 - jy0);
            float iw = fminf(x1, jx1) - fmaxf(x0, jx0);
            float ih = fminf(y1, jy1) - fmaxf(y0, jy0);
            iw = fmaxf(iw, 0.0f); ih = fmaxf(ih, 0.0f);
            float inter = iw * ih;
            float iou = inter / (ai + aj - inter + 1e-6f);
            if (iou > NMS_THR) bits |= (1u << t);
        }
    }
    mask[(size_t)i * 128 + wj] = bits;
}

// ---------------------------------------------------------------------------
// Sequential greedy NMS (matches jax fori_loop) in one workgroup; keep bitmask
// lives in LDS (128 words). The next suppression row is prefetched
// (global_prefetch_b8) while the current row is applied, hiding the 512B row
// fetch behind the two barriers of the serial dependence chain.
__global__ void __launch_bounds__(128) k_nms(const unsigned* __restrict__ mask,
                                             const float* __restrict__ tops,
                                             float* __restrict__ out) {
    __shared__ unsigned keep[128];
    int t = threadIdx.x;
    unsigned w = 0u;
    #pragma unroll 4
    for (int b = 0; b < 32; ++b)
        if (tops[t * 32 + b] > SCORE_THR) w |= (1u << b);
    keep[t] = w;
    __builtin_prefetch(&mask[t], 0, 3);
    __syncthreads();
    for (int i = 0; i < TOPK; ++i) {
        bool alive = (keep[i >> 5] >> (i & 31)) & 1u;
        if (i + 1 < TOPK)
            __builtin_prefetch(&mask[(size_t)(i + 1) * 128 + t], 0, 3);
        __syncthreads();
        if (alive) keep[t] &= ~mask[(size_t)i * 128 + t];
        __syncthreads();
    }
    unsigned kw = keep[t];
    #pragma unroll 4
    for (int b = 0; b < 32; ++b) {
        int idx = t * 32 + b;
        out[(size_t)idx * 25 + 24] = ((kw >> b) & 1u) ? tops[idx] : 0.0f;
    }
}

// ---------------------------------------------------------------------------
extern "C" void kernel_launch(void* const* d_in, const int* in_sizes, int n_in,
                              void* d_out, int out_size, void* d_ws, size_t ws_size,
                              hipStream_t stream) {
    const float* anchors = (const float*)d_in[0];
    const float* psm_v   = (const float*)d_in[1];
    const float* rm_v    = (const float*)d_in[2];
    const float* psm_i   = (const float*)d_in[3];
    const float* rm_i    = (const float*)d_in[4];
    const float* t_proj  = (const float*)d_in[5];
    const float* t_ego   = (const float*)d_in[6];
    float* out = (float*)d_out;                 // 4096 x 25 f32

    unsigned* ws32 = (unsigned*)d_ws;           // needs ~19.3 MB of workspace
    unsigned* hist   = ws32 + HIST_OFF;
    unsigned* cnt    = ws32 + CNT_OFF;
    unsigned* selBin = ws32 + BIN_OFF;
    unsigned long long* cand = (unsigned long long*)(ws32 + CAND_OFF);
    unsigned* ukeys  = ws32 + UKEY_OFF;
    float*    boxes  = (float*)(ws32 + BOXES_OFF);
    unsigned* topi   = ws32 + TOPI_OFF;
    float*    tops   = (float*)(ws32 + TOPS_OFF);
    float*    standup = (float*)(ws32 + STAND_OFF);
    unsigned* mask   = ws32 + MASK_OFF;

    k_zero<<<(ZERO_WORDS + 255) / 256, 256, 0, stream>>>(ws32);
    k_decode<<<NTOT / 256, 256, 0, stream>>>(anchors, psm_v, rm_v, psm_i, rm_i,
                                             t_proj, boxes, ukeys, hist);
    k_selectbin<<<1, 256, 0, stream>>>(hist, selBin);
    k_compact<<<NTOT / 256, 256, 0, stream>>>(ukeys, selBin, cnt, cand);
    k_sortcand<<<1, 1024, 0, stream>>>(cand, topi, tops);
    k_project<<<TOPK / 64, 256, 0, stream>>>(boxes, topi, t_ego, out, standup);
    k_mask<<<TOPK, 128, 0, stream>>>(standup, mask);
    k_nms<<<1, 128, 0, stream>>>(mask, tops, out);
}